// shift_34110630265566
// MI455X (gfx1250) — compile-verified
//
#include <hip/hip_runtime.h>

// Active-shift layer, (B=32, C=288, H=64, W=64) fp32.
// cpg = 32. Group g in [0,8): out[b,c,h,w] = x[b,c,h+dh,w+dw] (zero pad);
// channels 256..287 are all zeros.
//
// Pure streaming relocation: ~302 MB total traffic, 0 FLOPs -> HBM-bound,
// ~13 us floor at 23.3 TB/s. Design:
//   - one block per (b,c) plane (16 KB), c block-uniform -> all shift logic
//     in SALU, shift table packed into 2-bit immediates (no .rodata s_load)
//   - each thread: 4 independent NT load/store pairs (MLP, clause-able)
//   - dw=+-1 interior lanes use a 4B-aligned b128 load (legal per ISA 9.5:
//     DWORD mode only clears addr[1:0] for non-formatted B128); only the two
//     W-edge lanes per wave fall back to predicated dword loads
//   - NT hints both directions: every byte touched exactly once, keep the
//     302 MB stream out of L2's working set.

typedef float f32x4  __attribute__((ext_vector_type(4)));
typedef float f32x4u __attribute__((ext_vector_type(4), aligned(4)));

__global__ __launch_bounds__(256) void shift_34110630265566_kernel(
    const float* __restrict__ in, float* __restrict__ out) {
  constexpr int C = 288, H = 64, W = 64;

  const int tid = threadIdx.x;
  const int w  = (tid & 15) << 2;   // 0,4,...,60 (float4-aligned)
  const int h0 = tid >> 4;          // 0..15; thread covers h0 + {0,16,32,48}
  const int c = blockIdx.x;         // uniform per block
  const int b = blockIdx.y;         // uniform per block

  const int plane = (b * C + c) * (H * W);  // < 2^31: 32-bit addressing
  float* __restrict__ outcol = out + plane + w;

  const int g = c >> 5;             // 0..8
  if (g >= 8) {                     // block-uniform: zero group
    const f32x4 z = {0.f, 0.f, 0.f, 0.f};
#pragma unroll
    for (int k = 0; k < 4; ++k)
      __builtin_nontemporal_store(z, (f32x4*)(outcol + (h0 + 16 * k) * W));
    return;
  }

  // dh in {-1,1,0,0,-1,-1,1,1}, dw in {0,0,-1,1,-1,1,-1,1}, packed as
  // (val+1) in 2-bit fields -> pure SALU decode, no constant-memory load.
  const int dh = ((0xA058 >> (2 * g)) & 3) - 1;
  const int dw = ((0x8885 >> (2 * g)) & 3) - 1;

#pragma unroll
  for (int k = 0; k < 4; ++k) {
    const int h  = h0 + 16 * k;
    const int ih = h + dh;
    f32x4 v = {0.f, 0.f, 0.f, 0.f};
    if (ih >= 0 && ih < H) {        // diverges only for h==0 / h==63 lanes
      const float* __restrict__ row = in + plane + ih * W;
      if (dw == 0) {
        // Same W-phase: one aligned 16B NT load.
        v = __builtin_nontemporal_load((const f32x4*)(row + w));
      } else {
        const int wb = w + dw;      // first input index for this quad
        if (wb >= 0 && wb + 3 < W) {
          // Interior lanes (30 of 32): single 4B-aligned b128 NT load.
          v = __builtin_nontemporal_load((const f32x4u*)(row + wb));
        } else {
          // W-edge lanes (w==0 with dw<0, or w==60 with dw>0): predicated
          // dword loads; OOB element stays zero (matches zero padding).
#pragma unroll
          for (int i = 0; i < 4; ++i) {
            const int iw = wb + i;
            if (iw >= 0 && iw < W) {
              v[i] = __builtin_nontemporal_load(row + iw);
            }
          }
        }
      }
    }
    __builtin_nontemporal_store(v, (f32x4*)(outcol + h * W));
  }
}

extern "C" void kernel_launch(void* const* d_in, const int* in_sizes, int n_in,
                              void* d_out, int out_size, void* d_ws, size_t ws_size,
                              hipStream_t stream) {
  const float* x = (const float*)d_in[0];
  float* out = (float*)d_out;

  constexpr int C = 288, H = 64, W = 64;
  const int B = in_sizes[0] / (C * H * W);  // 32 for the reference shapes

  // One block per (b,c) plane; 256 threads x 4 float4 = 1024 float4 = H*W.
  dim3 grid(C, B);
  shift_34110630265566_kernel<<<grid, 256, 0, stream>>>(x, out);
}